// GATModel_18047452577827
// MI455X (gfx1250) — compile-verified
//
#include <hip/hip_runtime.h>
#include <hip/hip_bf16.h>

// ---------------------------------------------------------------------------
// GAT (2-layer, HEADS=2, OUT=32) for gfx1250 / MI455X.
// N=100000 (16*6250), hidden=64 (4*16), K1=128, K2=64 -> exact WMMA tiling.
// ---------------------------------------------------------------------------

typedef __attribute__((ext_vector_type(2))) float v2f;
typedef __attribute__((ext_vector_type(8))) float v8f;

#define GAT_N      100000
#define GAT_F      64          // HEADS*OUT
#define GAT_H      2
#define GAT_C      32          // OUT (per head)
#define NEG_SLOPE  0.2f

// ---- order-preserving float <-> uint encoding for atomic max -------------
__device__ __forceinline__ unsigned f2ord(float f) {
    unsigned u = __float_as_uint(f);
    return (u & 0x80000000u) ? ~u : (u | 0x80000000u);
}
__device__ __forceinline__ float ord2f(unsigned u) {
    unsigned b = (u & 0x80000000u) ? (u ^ 0x80000000u) : ~u;
    return __uint_as_float(b);
}

__device__ __forceinline__ void edge_sd(const int* __restrict__ ei, int e, int E,
                                        int& s, int& d) {
    if (e < E) { s = ei[e]; d = ei[E + e]; }
    else       { s = e - E; d = e - E; }        // appended self-loops
}

__device__ __forceinline__ float leaky(float x) {
    return x > 0.0f ? x : NEG_SLOPE * x;
}

// ---------------------------------------------------------------------------
// Dense projection  C[M x Ncols] = A[M x K] * B[K x Ncols]  via
// V_WMMA_F32_16X16X4_F32. One wave per 16x16 output tile, exact tiling
// (M % 16 == 0, Ncols % 16 == 0, K % 4 == 0), EXEC all-ones inside the wave.
// ---------------------------------------------------------------------------
__global__ void gat_gemm_wmma_f32(const float* __restrict__ A,
                                  const float* __restrict__ B,
                                  float* __restrict__ C,
                                  int M, int Ncols, int K) {
    const int wave  = (blockIdx.x * blockDim.x + threadIdx.x) >> 5;
    const int lane  = threadIdx.x & 31;
    const int ntile = Ncols >> 4;
    const int mt    = wave / ntile;
    const int nt    = wave - mt * ntile;
    if (mt * 16 >= M) return;                    // wave-uniform

    const int rowA = mt * 16 + (lane & 15);      // A: lane holds row M=lane&15
    const int kgrp = (lane >> 4) * 2;            // K = {0,1} lanes 0-15, {2,3} lanes 16-31
    const int colB = nt * 16 + (lane & 15);      // B: lane holds col N=lane&15

    const float* __restrict__ arow = A + (long long)rowA * K;
    v8f acc = {};
    for (int k0 = 0; k0 < K; k0 += 4) {
        v2f a, b;
        a.x = arow[k0 + kgrp];
        a.y = arow[k0 + kgrp + 1];
        b.x = B[(long long)(k0 + kgrp)     * Ncols + colB];
        b.y = B[(long long)(k0 + kgrp + 1) * Ncols + colB];
        acc = __builtin_amdgcn_wmma_f32_16x16x4_f32(
                  false, a, false, b, (short)0, acc, false, false);
    }
    // D/C layout: VGPR r -> row (mt*16 + r + 8*(lane>>4)), col = nt*16 + (lane&15)
    const int crow = mt * 16 + 8 * (lane >> 4);
    const int ccol = nt * 16 + (lane & 15);
#pragma unroll
    for (int r = 0; r < 8; ++r)
        C[(long long)(crow + r) * Ncols + ccol] = acc[r];
}

// ---------------------------------------------------------------------------
// Per-node attention logits: al[n,h] = dot(xp[n, h*32 : h*32+32], a[h,:])
// One wave per (node, head); wave32 shuffle reduction.
// ---------------------------------------------------------------------------
__global__ void gat_node_logits(const float* __restrict__ xp,
                                const float* __restrict__ a_src,
                                const float* __restrict__ a_dst,
                                float* __restrict__ al_src,
                                float* __restrict__ al_dst,
                                int n_nodes) {
    const int wid  = (blockIdx.x * blockDim.x + threadIdx.x) >> 5;
    const int lane = threadIdx.x & 31;
    if (wid >= n_nodes * GAT_H) return;
    const int node = wid >> 1;
    const int head = wid & 1;

    const float v  = xp[(long long)node * GAT_F + head * GAT_C + lane];
    float s1 = v * a_src[head * GAT_C + lane];
    float s2 = v * a_dst[head * GAT_C + lane];
#pragma unroll
    for (int off = 16; off > 0; off >>= 1) {
        s1 += __shfl_xor(s1, off, 32);
        s2 += __shfl_xor(s2, off, 32);
    }
    if (lane == 0) { al_src[wid] = s1; al_dst[wid] = s2; }
}

// ---------------------------------------------------------------------------
// Init per-layer state: m keys = 0 (== -NaN sentinel), s = 0,
// out[n, c] = bias[c]  (so the scatter-add lands on top of the bias).
// Launched with n_nodes*64 threads.
// ---------------------------------------------------------------------------
__global__ void gat_init_layer(unsigned* __restrict__ mkey,
                               float* __restrict__ ssum,
                               float* __restrict__ out,
                               const float* __restrict__ bias,
                               int n_nodes) {
    const int i = blockIdx.x * blockDim.x + threadIdx.x;
    if (i < n_nodes * GAT_H) { mkey[i] = 0u; ssum[i] = 0.0f; }
    if (i < n_nodes * GAT_F) out[i] = bias[i & (GAT_F - 1)];
}

// ---------------------------------------------------------------------------
// Pass A: segment max of leaky-relu logits (atomic max on ordered uint keys).
// One thread per (edge, head).
// ---------------------------------------------------------------------------
__global__ void gat_edge_max(const int* __restrict__ ei, int E, int Etot,
                             const float* __restrict__ al_s,
                             const float* __restrict__ al_d,
                             unsigned* __restrict__ mkey) {
    const int t = blockIdx.x * blockDim.x + threadIdx.x;
    if (t >= Etot * GAT_H) return;
    const int e = t >> 1, h = t & 1;
    int s, d; edge_sd(ei, e, E, s, d);
    const float lg = leaky(al_s[s * GAT_H + h] + al_d[d * GAT_H + h]);
    atomicMax(&mkey[d * GAT_H + h], f2ord(lg));
}

// ---------------------------------------------------------------------------
// Pass B: segment sum of exp(logit - m[dst]).
// ---------------------------------------------------------------------------
__global__ void gat_edge_sum(const int* __restrict__ ei, int E, int Etot,
                             const float* __restrict__ al_s,
                             const float* __restrict__ al_d,
                             const unsigned* __restrict__ mkey,
                             float* __restrict__ ssum) {
    const int t = blockIdx.x * blockDim.x + threadIdx.x;
    if (t >= Etot * GAT_H) return;
    const int e = t >> 1, h = t & 1;
    int s, d; edge_sd(ei, e, E, s, d);
    const float lg = leaky(al_s[s * GAT_H + h] + al_d[d * GAT_H + h]);
    const float m  = ord2f(mkey[d * GAT_H + h]);
    atomicAdd(&ssum[d * GAT_H + h], __expf(lg - m));
}

// ---------------------------------------------------------------------------
// Pass C: weighted scatter-aggregate. 16 threads per edge; thread q handles
// channels [4q, 4q+4) (float4 gather from xp[src], 4x global_atomic_add_f32
// into out[dst]). xp and out both live in the 192 MB L2 -> on-chip traffic.
// ---------------------------------------------------------------------------
__global__ void gat_edge_agg(const int* __restrict__ ei, int E, int Etot,
                             const float* __restrict__ al_s,
                             const float* __restrict__ al_d,
                             const unsigned* __restrict__ mkey,
                             const float* __restrict__ ssum,
                             const float* __restrict__ xp,
                             float* __restrict__ out) {
    const int t = blockIdx.x * blockDim.x + threadIdx.x;
    if (t >= Etot * 16) return;
    const int e = t >> 4;
    const int q = t & 15;          // quad of channels
    const int h = q >> 3;          // head = (4q)/32
    int s, d; edge_sd(ei, e, E, s, d);

    const float lg    = leaky(al_s[s * GAT_H + h] + al_d[d * GAT_H + h]);
    const float m     = ord2f(mkey[d * GAT_H + h]);
    const float denom = ssum[d * GAT_H + h] + 1e-16f;
    const float alpha = __expf(lg - m) / denom;

    const float4 xv = *(const float4*)(xp + (long long)s * GAT_F + q * 4);
    float* op = out + (long long)d * GAT_F + q * 4;
    atomicAdd(op + 0, xv.x * alpha);
    atomicAdd(op + 1, xv.y * alpha);
    atomicAdd(op + 2, xv.z * alpha);
    atomicAdd(op + 3, xv.w * alpha);
}

// ---------------------------------------------------------------------------
// ELU in place (bias already folded in by gat_init_layer).
// ---------------------------------------------------------------------------
__global__ void gat_elu(float* __restrict__ h, int total) {
    const int i = blockIdx.x * blockDim.x + threadIdx.x;
    if (i >= total) return;
    const float v = h[i];
    h[i] = v > 0.0f ? v : (__expf(v) - 1.0f);
}

// ---------------------------------------------------------------------------
extern "C" void kernel_launch(void* const* d_in, const int* in_sizes, int n_in,
                              void* d_out, int out_size, void* d_ws, size_t ws_size,
                              hipStream_t stream) {
    const float* x    = (const float*)d_in[0];
    const int*   ei   = (const int*)  d_in[1];
    const float* W1   = (const float*)d_in[2];
    const float* as1  = (const float*)d_in[3];
    const float* ad1  = (const float*)d_in[4];
    const float* b1   = (const float*)d_in[5];
    const float* W2   = (const float*)d_in[6];
    const float* as2  = (const float*)d_in[7];
    const float* ad2  = (const float*)d_in[8];
    const float* b2   = (const float*)d_in[9];
    float*       out  = (float*)d_out;

    const int N    = GAT_N;
    const int E    = in_sizes[1] / 2;     // edge_index is (2, E)
    const int Etot = E + N;               // + self loops

    // ---- workspace layout (floats) ----
    float*    ws   = (float*)d_ws;
    float*    xp   = ws;                          // N*64
    float*    h    = xp  + (size_t)N * GAT_F;     // N*64
    float*    als  = h   + (size_t)N * GAT_F;     // N*2
    float*    ald  = als + (size_t)N * GAT_H;     // N*2
    unsigned* mkey = (unsigned*)(ald + (size_t)N * GAT_H); // N*2
    float*    ssum = (float*)(mkey + (size_t)N * GAT_H);   // N*2

    const int TB = 256;
    const int g_gemm  = ((N / 16) * (GAT_F / 16) * 32 + TB - 1) / TB; // waves*32 threads
    const int g_log   = (N * GAT_H * 32 + TB - 1) / TB;
    const int g_init  = (N * GAT_F + TB - 1) / TB;
    const int g_eh    = (Etot * GAT_H + TB - 1) / TB;
    const int g_agg   = (Etot * 16 + TB - 1) / TB;
    const int g_elu   = (N * GAT_F + TB - 1) / TB;

    // ================= layer 1 =================
    gat_gemm_wmma_f32<<<g_gemm, TB, 0, stream>>>(x, W1, xp, N, GAT_F, 128);
    gat_node_logits <<<g_log,  TB, 0, stream>>>(xp, as1, ad1, als, ald, N);
    gat_init_layer  <<<g_init, TB, 0, stream>>>(mkey, ssum, h, b1, N);
    gat_edge_max    <<<g_eh,   TB, 0, stream>>>(ei, E, Etot, als, ald, mkey);
    gat_edge_sum    <<<g_eh,   TB, 0, stream>>>(ei, E, Etot, als, ald, mkey, ssum);
    gat_edge_agg    <<<g_agg,  TB, 0, stream>>>(ei, E, Etot, als, ald, mkey, ssum, xp, h);
    gat_elu         <<<g_elu,  TB, 0, stream>>>(h, N * GAT_F);

    // ================= layer 2 =================
    gat_gemm_wmma_f32<<<g_gemm, TB, 0, stream>>>(h, W2, xp, N, GAT_F, GAT_F);
    gat_node_logits <<<g_log,  TB, 0, stream>>>(xp, as2, ad2, als, ald, N);
    gat_init_layer  <<<g_init, TB, 0, stream>>>(mkey, ssum, out, b2, N);
    gat_edge_max    <<<g_eh,   TB, 0, stream>>>(ei, E, Etot, als, ald, mkey);
    gat_edge_sum    <<<g_eh,   TB, 0, stream>>>(ei, E, Etot, als, ald, mkey, ssum);
    gat_edge_agg    <<<g_agg,  TB, 0, stream>>>(ei, E, Etot, als, ald, mkey, ssum, xp, out);
}